// SpatialNeighborAttentionBlock_64072322122336
// MI455X (gfx1250) — compile-verified
//
#include <hip/hip_runtime.h>
#include <hip/hip_bf16.h>

#define E_N 10000
#define T_N 24
#define D_N 128
#define H_N 8
#define K_N 16
#define FF_N 512
#define EPS_LN 1e-5f

typedef float v2f __attribute__((ext_vector_type(2)));
typedef float v8f __attribute__((ext_vector_type(8)));

// D = A(16x4, f32) * B(4x16, f32) + C(16x16, f32)  -- exact fp32 WMMA on gfx1250
__device__ __forceinline__ v8f wmma_f32(v2f a, v2f b, v8f c) {
  return __builtin_amdgcn_wmma_f32_16x16x4_f32(
      /*neg_a=*/false, a, /*neg_b=*/false, b,
      /*c_mod=*/(short)0, c, /*reuse_a=*/false, /*reuse_b=*/false);
}

// Computes four 16x16 tiles of C = A(32xK) * B(KxN_total):
//   acc[0]: rows 0..15,  cols wave*32+0..15
//   acc[1]: rows 0..15,  cols wave*32+16..31
//   acc[2]: rows 16..31, cols wave*32+0..15
//   acc[3]: rows 16..31, cols wave*32+16..31
// A in LDS (row stride 128 floats), B row-major in global (row stride ldb).
__device__ __forceinline__ void gemm_acc(const float* sA, const float* B, int ldb,
                                         int wave, int half, int lm, int Kdim,
                                         v8f acc[4]) {
  const int n0 = wave * 32 + lm;
  const int n1 = n0 + 16;
  for (int k0 = 0; k0 < Kdim; k0 += 4) {
    const int ka = k0 + 2 * half;
    v2f a0, a1, b0, b1;
    a0.x = sA[lm * 128 + ka];        a0.y = sA[lm * 128 + ka + 1];
    a1.x = sA[(16 + lm) * 128 + ka]; a1.y = sA[(16 + lm) * 128 + ka + 1];
    b0.x = B[(size_t)ka * ldb + n0]; b0.y = B[(size_t)(ka + 1) * ldb + n0];
    b1.x = B[(size_t)ka * ldb + n1]; b1.y = B[(size_t)(ka + 1) * ldb + n1];
    acc[0] = wmma_f32(a0, b0, acc[0]);
    acc[1] = wmma_f32(a0, b1, acc[1]);
    acc[2] = wmma_f32(a1, b0, acc[2]);
    acc[3] = wmma_f32(a1, b1, acc[3]);
  }
}

// ---------------- Kernel 1: Kx = x0 @ Wk, Vx = x0 @ Wv for all events --------
__global__ __launch_bounds__(128) void kv_proj_kernel(
    const float* __restrict__ x, const float* __restrict__ Wk,
    const float* __restrict__ Wv, float* __restrict__ Kx, float* __restrict__ Vx) {
  __shared__ float sx0[16 * 128];
  const int tid = threadIdx.x;
  const int e0 = blockIdx.x * 16;
  for (int i = tid; i < 16 * 128; i += 128) {
    const int r = i >> 7, c = i & 127;
    sx0[i] = x[(size_t)(e0 + r) * (T_N * D_N) + c];  // x[e][0][c]
  }
  __syncthreads();

  const int lane = tid & 31, wave = tid >> 5;
  const int half = lane >> 4, lm = lane & 15;
  const int n0 = wave * 32 + lm, n1 = n0 + 16;
  v8f aK0{}, aK1{}, aV0{}, aV1{};
  for (int k0 = 0; k0 < 128; k0 += 4) {
    const int ka = k0 + 2 * half;
    v2f a, bk0, bk1, bv0, bv1;
    a.x = sx0[lm * 128 + ka];  a.y = sx0[lm * 128 + ka + 1];
    bk0.x = Wk[ka * 128 + n0]; bk0.y = Wk[(ka + 1) * 128 + n0];
    bk1.x = Wk[ka * 128 + n1]; bk1.y = Wk[(ka + 1) * 128 + n1];
    bv0.x = Wv[ka * 128 + n0]; bv0.y = Wv[(ka + 1) * 128 + n0];
    bv1.x = Wv[ka * 128 + n1]; bv1.y = Wv[(ka + 1) * 128 + n1];
    aK0 = wmma_f32(a, bk0, aK0);
    aK1 = wmma_f32(a, bk1, aK1);
    aV0 = wmma_f32(a, bv0, aV0);
    aV1 = wmma_f32(a, bv1, aV1);
  }
  for (int r = 0; r < 8; ++r) {
    const size_t row = (size_t)(e0 + r + 8 * half) * 128;
    Kx[row + n0] = aK0[r];
    Kx[row + n1] = aK1[r];
    Vx[row + n0] = aV0[r];
    Vx[row + n1] = aV1[r];
  }
}

// ---------------- Kernel 2: fused attention + FFN block per event ------------
__global__ __launch_bounds__(128) void attn_block_kernel(
    const float* __restrict__ x, const int* __restrict__ nbr,
    const float* __restrict__ Wq, const float* __restrict__ Wo,
    const float* __restrict__ bo, const float* __restrict__ g1,
    const float* __restrict__ b1, const float* __restrict__ g2,
    const float* __restrict__ b2, const float* __restrict__ W1,
    const float* __restrict__ bf1, const float* __restrict__ W2,
    const float* __restrict__ bf2, const float* __restrict__ Kx,
    const float* __restrict__ Vx, float* __restrict__ out) {
  __shared__ float buf0[32 * 128];  // x (rows 24..31 zero)
  __shared__ float buf1[32 * 128];  // Q -> Z -> FFN hidden G -> pre-LN2
  __shared__ float buf2[32 * 128];  // Kn(rows0-15)+Vn(rows16-31) -> pre-LN1
  __shared__ float buf3[32 * 128];  // attn scores -> h (post-LN1)
  const int VOFF = 16 * 128;

  const int tid = threadIdx.x;
  const int e = blockIdx.x;
  const int lane = tid & 31, wave = tid >> 5;
  const int half = lane >> 4, lm = lane & 15;
  const size_t xbase = (size_t)e * (T_N * D_N);

  // stage x (zero-pad rows) and gather neighbor K/V rows
  for (int i = tid; i < 32 * 128; i += 128) {
    const int r = i >> 7, c = i & 127;
    buf0[i] = (r < T_N) ? x[xbase + (size_t)r * D_N + c] : 0.0f;
  }
  for (int i = tid; i < 16 * 128; i += 128) {
    const int k = i >> 7, c = i & 127;
    const int n = nbr[e * K_N + k];
    buf2[i]        = Kx[(size_t)n * D_N + c];
    buf2[VOFF + i] = Vx[(size_t)n * D_N + c];
  }
  __syncthreads();

  // ---- Q = (x @ Wq) * rsqrt(DH) -> buf1 ----
  {
    v8f q[4] = {};
    gemm_acc(buf0, Wq, 128, wave, half, lm, 128, q);
    for (int i = 0; i < 4; ++i)
      for (int r = 0; r < 8; ++r) {
        const int row = ((i >> 1) << 4) + r + 8 * half;
        const int col = wave * 32 + ((i & 1) << 4) + lm;
        buf1[row * 128 + col] = q[i][r] * 0.25f;  // 1/sqrt(16)
      }
  }
  __syncthreads();

  // ---- attention scores: per head 24x16 @ 16x16 (K=DH=16) -> buf3 ----
  {
    const int hc0 = (2 * wave) * 16, hc1 = hc0 + 16;  // this wave's two heads
    v8f s0{}, s1{}, s2{}, s3{};
    for (int k0 = 0; k0 < 16; k0 += 4) {
      const int ka = k0 + 2 * half;
      v2f a00, a01, a10, a11, b0, b1;
      a00.x = buf1[lm * 128 + hc0 + ka];        a00.y = buf1[lm * 128 + hc0 + ka + 1];
      a01.x = buf1[(16 + lm) * 128 + hc0 + ka]; a01.y = buf1[(16 + lm) * 128 + hc0 + ka + 1];
      a10.x = buf1[lm * 128 + hc1 + ka];        a10.y = buf1[lm * 128 + hc1 + ka + 1];
      a11.x = buf1[(16 + lm) * 128 + hc1 + ka]; a11.y = buf1[(16 + lm) * 128 + hc1 + ka + 1];
      // B[d][k] = Kn[k][head*16+d]  (col index = neighbor k = lm)
      b0.x = buf2[lm * 128 + hc0 + ka]; b0.y = buf2[lm * 128 + hc0 + ka + 1];
      b1.x = buf2[lm * 128 + hc1 + ka]; b1.y = buf2[lm * 128 + hc1 + ka + 1];
      s0 = wmma_f32(a00, b0, s0);
      s1 = wmma_f32(a01, b0, s1);
      s2 = wmma_f32(a10, b1, s2);
      s3 = wmma_f32(a11, b1, s3);
    }
    for (int r = 0; r < 8; ++r) {
      const int t0 = r + 8 * half, t1 = 16 + r + 8 * half;
      buf3[t0 * 128 + hc0 + lm] = s0[r];
      buf3[t1 * 128 + hc0 + lm] = s1[r];
      buf3[t0 * 128 + hc1 + lm] = s2[r];
      buf3[t1 * 128 + hc1 + lm] = s3[r];
    }
  }
  __syncthreads();

  // ---- softmax over K=16 neighbors, rows t<24, all heads ----
  for (int idx = tid; idx < T_N * H_N; idx += 128) {
    const int t = idx >> 3, h = idx & 7;
    float* row = &buf3[t * 128 + h * 16];
    float m = row[0];
    for (int k = 1; k < 16; ++k) m = fmaxf(m, row[k]);
    float s = 0.f;
    for (int k = 0; k < 16; ++k) { row[k] = __expf(row[k] - m); s += row[k]; }
    const float inv = 1.f / s;
    for (int k = 0; k < 16; ++k) row[k] *= inv;
  }
  __syncthreads();

  // ---- Z = attn @ Vn (per head, K=16) -> buf1 ----
  {
    const int hc0 = (2 * wave) * 16, hc1 = hc0 + 16;
    v8f z0{}, z1{}, z2{}, z3{};
    for (int k0 = 0; k0 < 16; k0 += 4) {
      const int ka = k0 + 2 * half;
      v2f a00, a01, a10, a11, b0, b1;
      a00.x = buf3[lm * 128 + hc0 + ka];        a00.y = buf3[lm * 128 + hc0 + ka + 1];
      a01.x = buf3[(16 + lm) * 128 + hc0 + ka]; a01.y = buf3[(16 + lm) * 128 + hc0 + ka + 1];
      a10.x = buf3[lm * 128 + hc1 + ka];        a10.y = buf3[lm * 128 + hc1 + ka + 1];
      a11.x = buf3[(16 + lm) * 128 + hc1 + ka]; a11.y = buf3[(16 + lm) * 128 + hc1 + ka + 1];
      // B[k][d] = Vn[k][head*16+d]
      b0.x = buf2[VOFF + ka * 128 + hc0 + lm]; b0.y = buf2[VOFF + (ka + 1) * 128 + hc0 + lm];
      b1.x = buf2[VOFF + ka * 128 + hc1 + lm]; b1.y = buf2[VOFF + (ka + 1) * 128 + hc1 + lm];
      z0 = wmma_f32(a00, b0, z0);
      z1 = wmma_f32(a01, b0, z1);
      z2 = wmma_f32(a10, b1, z2);
      z3 = wmma_f32(a11, b1, z3);
    }
    for (int r = 0; r < 8; ++r) {
      const int t0 = r + 8 * half, t1 = 16 + r + 8 * half;
      buf1[t0 * 128 + hc0 + lm] = z0[r];
      buf1[t1 * 128 + hc0 + lm] = z1[r];
      buf1[t0 * 128 + hc1 + lm] = z2[r];
      buf1[t1 * 128 + hc1 + lm] = z3[r];
    }
  }
  __syncthreads();

  // ---- pre-LN1 = x + Z @ Wo + bo -> buf2 ----
  {
    v8f o[4] = {};
    gemm_acc(buf1, Wo, 128, wave, half, lm, 128, o);
    for (int i = 0; i < 4; ++i)
      for (int r = 0; r < 8; ++r) {
        const int row = ((i >> 1) << 4) + r + 8 * half;
        const int col = wave * 32 + ((i & 1) << 4) + lm;
        buf2[row * 128 + col] = o[i][r] + bo[col] + buf0[row * 128 + col];
      }
  }
  __syncthreads();

  // ---- LN1 -> h in buf3 (rows >=24 untouched: they hold zeros) ----
  if (tid < T_N) {
    const float* rp = &buf2[tid * 128];
    float mu = 0.f;
    for (int c = 0; c < 128; ++c) mu += rp[c];
    mu *= (1.f / 128.f);
    float var = 0.f;
    for (int c = 0; c < 128; ++c) { const float d = rp[c] - mu; var += d * d; }
    var *= (1.f / 128.f);
    const float rs = rsqrtf(var + EPS_LN);
    float* wp = &buf3[tid * 128];
    for (int c = 0; c < 128; ++c) wp[c] = (rp[c] - mu) * rs * g1[c] + b1[c];
  }
  __syncthreads();

  // ---- FFN in 4 column-chunks of W1; y accumulated in VGPRs ----
  v8f y[4] = {};
  for (int ch = 0; ch < 4; ++ch) {
    v8f g[4] = {};
    gemm_acc(buf3, W1 + ch * 128, FF_N, wave, half, lm, 128, g);
    for (int i = 0; i < 4; ++i)
      for (int r = 0; r < 8; ++r) {
        const int row = ((i >> 1) << 4) + r + 8 * half;
        const int col = wave * 32 + ((i & 1) << 4) + lm;
        buf1[row * 128 + col] = fmaxf(g[i][r] + bf1[ch * 128 + col], 0.f);
      }
    __syncthreads();
    gemm_acc(buf1, W2 + (size_t)(ch * 128) * D_N, 128, wave, half, lm, 128, y);
    __syncthreads();
  }

  // ---- pre-LN2 = h + y + bf2 -> buf1 ----
  for (int i = 0; i < 4; ++i)
    for (int r = 0; r < 8; ++r) {
      const int row = ((i >> 1) << 4) + r + 8 * half;
      const int col = wave * 32 + ((i & 1) << 4) + lm;
      buf1[row * 128 + col] = y[i][r] + bf2[col] + buf3[row * 128 + col];
    }
  __syncthreads();

  // ---- LN2 -> global out ----
  if (tid < T_N) {
    const float* rp = &buf1[tid * 128];
    float mu = 0.f;
    for (int c = 0; c < 128; ++c) mu += rp[c];
    mu *= (1.f / 128.f);
    float var = 0.f;
    for (int c = 0; c < 128; ++c) { const float d = rp[c] - mu; var += d * d; }
    var *= (1.f / 128.f);
    const float rs = rsqrtf(var + EPS_LN);
    for (int c = 0; c < 128; ++c)
      out[xbase + (size_t)tid * D_N + c] = (rp[c] - mu) * rs * g2[c] + b2[c];
  }
}

extern "C" void kernel_launch(void* const* d_in, const int* in_sizes, int n_in,
                              void* d_out, int out_size, void* d_ws, size_t ws_size,
                              hipStream_t stream) {
  const float* x   = (const float*)d_in[0];
  const int*   nbr = (const int*)d_in[1];
  const float* Wq  = (const float*)d_in[2];
  const float* Wk  = (const float*)d_in[3];
  const float* Wv  = (const float*)d_in[4];
  const float* Wo  = (const float*)d_in[5];
  const float* bo  = (const float*)d_in[6];
  const float* g1  = (const float*)d_in[7];
  const float* b1  = (const float*)d_in[8];
  const float* g2  = (const float*)d_in[9];
  const float* b2  = (const float*)d_in[10];
  const float* W1  = (const float*)d_in[11];
  const float* bf1 = (const float*)d_in[12];
  const float* W2  = (const float*)d_in[13];
  const float* bf2 = (const float*)d_in[14];

  float* Kx = (float*)d_ws;                       // E*D floats
  float* Vx = Kx + (size_t)E_N * D_N;             // E*D floats (10.24 MB total)

  kv_proj_kernel<<<E_N / 16, 128, 0, stream>>>(x, Wk, Wv, Kx, Vx);
  attn_block_kernel<<<E_N, 128, 0, stream>>>(x, nbr, Wq, Wo, bo, g1, b1, g2, b2,
                                             W1, bf1, W2, bf2, Kx, Vx,
                                             (float*)d_out);
}